// IW_MaxSquareloss_86517821215225
// MI455X (gfx1250) — compile-verified
//
#include <hip/hip_runtime.h>
#include <hip/hip_bf16.h>
#include <math.h>

#define NUM_CLASS 19
#define N_BATCH   4
#define HW        (512 * 1024)
#define RATIO_F   0.2f

typedef float v2f __attribute__((ext_vector_type(2)));
typedef float v4f __attribute__((ext_vector_type(4)));
typedef float v8f __attribute__((ext_vector_type(8)));

// ---------------------------------------------------------------------------
// Pass 0: zero the 2*76-float accumulator region in workspace (graph-safe).
// ---------------------------------------------------------------------------
__global__ void zero_ws_kernel(float* ws) {
    int i = threadIdx.x;
    if (i < 2 * N_BATCH * NUM_CLASS) ws[i] = 0.0f;
}

// ---------------------------------------------------------------------------
// Pass 1: stream prob once (NT b128 loads), per-pixel argmax + sum of squares,
// LDS-privatized histogram + per-class sumsq, then global float atomics.
// Memory-bound: 159 MB @ 23.3 TB/s ~ 7 us floor.
// ---------------------------------------------------------------------------
__global__ __launch_bounds__(256)
void hist_pass_kernel(const float* __restrict__ prob,
                      float* __restrict__ g_hist, float* __restrict__ g_S) {
    __shared__ float s_hist[NUM_CLASS];
    __shared__ float s_S[NUM_CLASS];

    const int t = threadIdx.x;
    if (t < NUM_CLASS) { s_hist[t] = 0.0f; s_S[t] = 0.0f; }
    __syncthreads();

    const int n = blockIdx.y;
    const float* __restrict__ pn = prob + (size_t)n * NUM_CLASS * HW;

    const int CHUNKS = 2;                         // vec4 pixels per thread
    int p = blockIdx.x * (blockDim.x * CHUNKS) + t;   // vec4 index in [0, HW/4)

    #pragma unroll
    for (int it = 0; it < CHUNKS; ++it, p += 256) {
        const size_t byteoff = (size_t)p * 4;
        // channel 0
        v4f v = __builtin_nontemporal_load((const v4f*)(pn + byteoff));
        v4f mx = v;
        v4f ss = v * v;
        int a0 = 0, a1 = 0, a2 = 0, a3 = 0;
        #pragma unroll
        for (int c = 1; c < NUM_CLASS; ++c) {
            v4f u = __builtin_nontemporal_load(
                        (const v4f*)(pn + (size_t)c * HW + byteoff));
            ss += u * u;
            // strict > keeps first occurrence (jnp.argmax semantics)
            if (u.x > mx.x) { mx.x = u.x; a0 = c; }
            if (u.y > mx.y) { mx.y = u.y; a1 = c; }
            if (u.z > mx.z) { mx.z = u.z; a2 = c; }
            if (u.w > mx.w) { mx.w = u.w; a3 = c; }
        }
        // mask = (maxpred != IGNORE_INDEX); masked pixels skip hist and loss
        if (mx.x != -1.0f) { atomicAdd(&s_hist[a0], 1.0f); atomicAdd(&s_S[a0], ss.x); }
        if (mx.y != -1.0f) { atomicAdd(&s_hist[a1], 1.0f); atomicAdd(&s_S[a1], ss.y); }
        if (mx.z != -1.0f) { atomicAdd(&s_hist[a2], 1.0f); atomicAdd(&s_S[a2], ss.z); }
        if (mx.w != -1.0f) { atomicAdd(&s_hist[a3], 1.0f); atomicAdd(&s_S[a3], ss.w); }
    }
    __syncthreads();

    if (t < NUM_CLASS) {
        atomicAdd(&g_hist[n * NUM_CLASS + t], s_hist[t]);
        atomicAdd(&g_S[n * NUM_CLASS + t],    s_S[t]);
    }
}

// ---------------------------------------------------------------------------
// Pass 2: one wave32. Compute per-(n,c) term = S / max(hist^0.2 * total^0.8, 1)
// and reduce the 76 terms with V_WMMA_F32_16X16X4_F32 (A = terms in the ISA
// 16x4 f32 layout, B = ones; two accumulating WMMAs cover 128 slots; column 0
// of D holds the row sums).
// ---------------------------------------------------------------------------
__global__ __launch_bounds__(32)
void finalize_kernel(const float* __restrict__ g_hist,
                     const float* __restrict__ g_S,
                     float* __restrict__ out) {
    __shared__ float s_total[N_BATCH];
    const int t = threadIdx.x;   // 0..31

    if (t < N_BATCH) {
        float tot = 0.0f;
        for (int c = 0; c < NUM_CLASS; ++c) tot += g_hist[t * NUM_CLASS + c];
        s_total[t] = tot;
    }
    __syncthreads();

    auto term = [&](int idx) -> float {
        if (idx >= N_BATCH * NUM_CLASS) return 0.0f;
        float h     = g_hist[idx];
        float tot   = s_total[idx / NUM_CLASS];
        float denom = fmaxf(__powf(h, RATIO_F) * __powf(tot, 1.0f - RATIO_F), 1.0f);
        return g_S[idx] / denom;
    };

    // 16x4 f32 A-matrix layout: lanes 0-15 hold (M=lane, K=0,1) in v[0],v[1];
    // lanes 16-31 hold (M=lane-16, K=2,3). Flat slot index = M*4 + K.
    const int m     = t & 15;
    const int kbase = (t < 16) ? 0 : 2;
    const int i0    = m * 4 + kbase;

    v2f A0; A0.x = term(i0);      A0.y = term(i0 + 1);
    v2f A1; A1.x = term(64 + i0); A1.y = term(64 + i0 + 1);
    v2f B;  B.x = 1.0f;           B.y = 1.0f;
    v8f C = {};

    float lane_sum;
#if defined(__has_builtin) && __has_builtin(__builtin_amdgcn_wmma_f32_16x16x4_f32)
    // D = A0*ones + (A1*ones + 0): every column of D replicates the row sums.
    C = __builtin_amdgcn_wmma_f32_16x16x4_f32(false, A1, false, B, (short)0, C,
                                              false, false);
    C = __builtin_amdgcn_wmma_f32_16x16x4_f32(false, A0, false, B, (short)0, C,
                                              false, false);
    // Lane L<16: v0..v7 = D[M=0..7][N=L];  lane L>=16: D[M=8..15][N=L-16].
    lane_sum = C[0] + C[1] + C[2] + C[3] + C[4] + C[5] + C[6] + C[7];
    lane_sum += __shfl_xor(lane_sum, 16, 32);   // rows 0-7 + rows 8-15
#else
    // Fallback: plain wave32 shuffle reduction of the 4 terms per lane.
    lane_sum = A0.x + A0.y + A1.x + A1.y;
    for (int d = 16; d > 0; d >>= 1) lane_sum += __shfl_xor(lane_sum, d, 32);
#endif

    if (t == 0) out[0] = -lane_sum / (float)(N_BATCH * NUM_CLASS);
}

// ---------------------------------------------------------------------------
extern "C" void kernel_launch(void* const* d_in, const int* in_sizes, int n_in,
                              void* d_out, int out_size, void* d_ws, size_t ws_size,
                              hipStream_t stream) {
    (void)in_sizes; (void)n_in; (void)out_size; (void)ws_size;
    // inputs: d_in[0] = pred (unused by the reference), d_in[1] = prob (f32)
    const float* prob = (const float*)d_in[1];

    float* g_hist = (float*)d_ws;                       // [4*19]
    float* g_S    = g_hist + N_BATCH * NUM_CLASS;       // [4*19]

    zero_ws_kernel<<<1, 256, 0, stream>>>((float*)d_ws);

    // HW/4 = 131072 vec4 pixels per sample; 512 vec4 per block -> 256 blocks/n
    dim3 grid(HW / 4 / (256 * 2), N_BATCH);
    hist_pass_kernel<<<grid, 256, 0, stream>>>(prob, g_hist, g_S);

    finalize_kernel<<<1, 32, 0, stream>>>(g_hist, g_S, (float*)d_out);
}